// VisualCortexV2_28638841930387
// MI455X (gfx1250) — compile-verified
//
#include <hip/hip_runtime.h>
#include <hip/hip_bf16.h>
#include <math.h>

typedef float v2f __attribute__((ext_vector_type(2)));
typedef float v8f __attribute__((ext_vector_type(8)));

#define BATCH   1024
#define HID     2048
#define IMGD    784
#define XW      794      // IMG_DIM + LBL_DIM
#define KTOP    102      // int(2048 * 0.05)

// ---------------------------------------------------------------------------
// 1) Row normalization: imgN[b, 0:784] = x[b,0:784] * 30 / (||row|| + 1e-8)
// ---------------------------------------------------------------------------
__global__ void __launch_bounds__(256) norm_kernel(const float* __restrict__ x,
                                                   float* __restrict__ imgN) {
    __shared__ float red[256];
    const int row = blockIdx.x;
    const float* xr = x + (size_t)row * XW;
    float s = 0.f;
    for (int j = threadIdx.x; j < IMGD; j += 256) { float f = xr[j]; s += f * f; }
    red[threadIdx.x] = s;
    __syncthreads();
    for (int off = 128; off > 0; off >>= 1) {
        if (threadIdx.x < off) red[threadIdx.x] += red[threadIdx.x + off];
        __syncthreads();
    }
    const float scale = 30.0f / (sqrtf(red[0]) + 1e-8f);
    for (int j = threadIdx.x; j < IMGD; j += 256)
        imgN[(size_t)row * IMGD + j] = xr[j] * scale;
}

// ---------------------------------------------------------------------------
// 2) labc[i][b][h] = sum_d lbl[b][d] * Li[h][d] + 2.0   (tiny K=10 GEMM)
// ---------------------------------------------------------------------------
__global__ void __launch_bounds__(256) labc_kernel(const float* __restrict__ x,
                                                   const float* __restrict__ L1,
                                                   const float* __restrict__ L2,
                                                   const float* __restrict__ L3,
                                                   float* __restrict__ labc) {
    const size_t idx = (size_t)blockIdx.x * 256 + threadIdx.x;   // < 3*B*H
    const int i  = (int)(idx / ((size_t)BATCH * HID));
    const size_t rem = idx % ((size_t)BATCH * HID);
    const int b = (int)(rem / HID);
    const int h = (int)(rem % HID);
    const float* L  = (i == 0) ? L1 : (i == 1 ? L2 : L3);
    const float* lb = x + (size_t)b * XW + IMGD;
    const float* Lr = L + (size_t)h * 10;
    float s = 2.0f;
#pragma unroll
    for (int d = 0; d < 10; ++d) s += lb[d] * Lr[d];
    labc[idx] = s;
}

// ---------------------------------------------------------------------------
// 3) WMMA GEMM with fused LIF epilogue:
//    cur = prev @ W^T ;  mems = 0.8*mems + 0.2*cur + labc
//    Wave tile: 32 (b) x 64 (h): two A fragments share each B fragment.
//    Explicit double-buffered K-chunks keep a full chunk of loads in flight
//    ahead of the 8 WMMAs consuming the previous chunk.
//    Block: 8 waves = 2 (b) x 4 (h) -> 64 x 256 tile.
//    Grid: (HID/256, BATCH/64) = (8, 16).
// ---------------------------------------------------------------------------
template <int K>
__global__ void __launch_bounds__(256) gemm_lif_kernel(const float* __restrict__ prev,
                                                       const float* __restrict__ W,
                                                       const float* __restrict__ labc,
                                                       float* __restrict__ mems) {
    const int lane = threadIdx.x & 31;
    const int wave = threadIdx.x >> 5;
    const int r    = lane & 15;
    const int half = lane >> 4;
    const int wb   = wave >> 2;          // 0..1 : b sub-tile (32 rows each)
    const int wh   = wave & 3;           // 0..3 : h column group (64 cols each)
    const int b_base = blockIdx.y * 64 + wb * 32;
    const int h_base = blockIdx.x * 256 + wh * 64;

    // A fragments: lane holds prev[b + r][k0 + 2*half + {0,1}]
    const float* Ap0 = prev + (size_t)(b_base + r) * K + 2 * half;
    const float* Ap1 = Ap0 + (size_t)16 * K;
    // B fragment t: lane holds W[h_base + 16*t + r][k0 + 2*half + {0,1}]
    const float* Bp = W + (size_t)(h_base + r) * K + 2 * half;
    const size_t bstride = (size_t)16 * K;

    v8f c0[4] = {v8f{}, v8f{}, v8f{}, v8f{}};
    v8f c1[4] = {v8f{}, v8f{}, v8f{}, v8f{}};

    // ---- software pipeline: prologue loads for chunk k0 = 0 ----
    v2f a0 = *(const v2f*)(Ap0);
    v2f a1 = *(const v2f*)(Ap1);
    v2f bf[4];
#pragma unroll
    for (int t = 0; t < 4; ++t) bf[t] = *(const v2f*)(Bp + (size_t)t * bstride);

#pragma unroll 2
    for (int k0 = 0; k0 < K - 4; k0 += 4) {
        // prefetch next chunk (no dependence on current WMMAs)
        v2f na0 = *(const v2f*)(Ap0 + k0 + 4);
        v2f na1 = *(const v2f*)(Ap1 + k0 + 4);
        v2f nbf[4];
#pragma unroll
        for (int t = 0; t < 4; ++t)
            nbf[t] = *(const v2f*)(Bp + (size_t)t * bstride + k0 + 4);

        // consume current chunk: 8 WMMAs
#pragma unroll
        for (int t = 0; t < 4; ++t) {
            c0[t] = __builtin_amdgcn_wmma_f32_16x16x4_f32(
                false, a0, false, bf[t], (short)0, c0[t], false, false);
            c1[t] = __builtin_amdgcn_wmma_f32_16x16x4_f32(
                false, a1, false, bf[t], (short)0, c1[t], false, false);
        }

        a0 = na0; a1 = na1;
#pragma unroll
        for (int t = 0; t < 4; ++t) bf[t] = nbf[t];
    }

    // ---- final chunk ----
#pragma unroll
    for (int t = 0; t < 4; ++t) {
        c0[t] = __builtin_amdgcn_wmma_f32_16x16x4_f32(
            false, a0, false, bf[t], (short)0, c0[t], false, false);
        c1[t] = __builtin_amdgcn_wmma_f32_16x16x4_f32(
            false, a1, false, bf[t], (short)0, c1[t], false, false);
    }

    // Epilogue: D layout -> c[v] is (b = b_tile + v + 8*half, h = h_base + 16*t + r)
#pragma unroll
    for (int t = 0; t < 4; ++t) {
#pragma unroll
        for (int v = 0; v < 8; ++v) {
            const int hh = h_base + 16 * t + r;
            {
                const int bb = b_base + v + 8 * half;
                const size_t idx = (size_t)bb * HID + hh;
                mems[idx] = mems[idx] * 0.8f + 0.2f * c0[t][v] + labc[idx];
            }
            {
                const int bb = b_base + 16 + v + 8 * half;
                const size_t idx = (size_t)bb * HID + hh;
                mems[idx] = mems[idx] * 0.8f + 0.2f * c1[t][v] + labc[idx];
            }
        }
    }
}

// ---------------------------------------------------------------------------
// 4) Exact per-row k-WTA (102nd largest) via 4-pass radix select on monotonic
//    keys, then fused mask / spike bonus / ReLU / trace EMA.
//    One block (256 threads) per row; 8 values per thread stay in registers.
// ---------------------------------------------------------------------------
__device__ __forceinline__ unsigned f2k(float f) {
    unsigned u = __float_as_uint(f);
    return (u & 0x80000000u) ? ~u : (u | 0x80000000u);   // monotonic increasing
}

__global__ void __launch_bounds__(256) topk_lif_kernel(float* __restrict__ mem,
                                                       float* __restrict__ act,
                                                       float* __restrict__ trace) {
    __shared__ unsigned hist[256];
    __shared__ unsigned s_prefix;
    __shared__ int s_kr;

    const int row = blockIdx.x;
    const int tid = threadIdx.x;
    float* m = mem + (size_t)row * HID;

    float v[8];
#pragma unroll
    for (int j = 0; j < 8; ++j) v[j] = m[tid + j * 256];

    if (tid == 0) { s_prefix = 0u; s_kr = KTOP; }
    __syncthreads();

#pragma unroll
    for (int p = 0; p < 4; ++p) {
        const int shift = 24 - 8 * p;
        hist[tid] = 0u;
        __syncthreads();
        const unsigned pref = s_prefix;
        const unsigned highmask = (p == 0) ? 0u : (0xFFFFFFFFu << (shift + 8));
#pragma unroll
        for (int j = 0; j < 8; ++j) {
            unsigned u = f2k(v[j]);
            if (((u ^ pref) & highmask) == 0u)
                atomicAdd(&hist[(u >> shift) & 255u], 1u);
        }
        __syncthreads();
        if (tid == 0) {
            unsigned acc = 0;
            int kr = s_kr;
            for (int bin = 255; bin >= 0; --bin) {
                unsigned cnt = hist[bin];
                if (acc + cnt >= (unsigned)kr) {
                    s_prefix = pref | ((unsigned)bin << shift);
                    s_kr = kr - (int)acc;
                    break;
                }
                acc += cnt;
            }
        }
        __syncthreads();
    }

    const unsigned thrK = s_prefix;   // exact key of the k-th largest element
    float* arow = act + (size_t)row * HID;
    float* trow = trace + (size_t)row * HID;
#pragma unroll
    for (int j = 0; j < 8; ++j) {
        const int col = tid + j * 256;
        const bool win = (f2k(v[j]) >= thrK);
        const float mv = win ? (v[j] + 0.5f) : 0.0f;   // mem*mask + 0.5*mask
        const float a  = fmaxf(mv, 0.0f);              // relu
        m[col]    = mv;
        arow[col] = a;
        trow[col] = 0.8f * trow[col] + 0.2f * a;
    }
}

// ---------------------------------------------------------------------------
extern "C" void kernel_launch(void* const* d_in, const int* in_sizes, int n_in,
                              void* d_out, int out_size, void* d_ws, size_t ws_size,
                              hipStream_t stream) {
    const float* x  = (const float*)d_in[0];
    const float* W1 = (const float*)d_in[1];
    const float* W2 = (const float*)d_in[2];
    const float* W3 = (const float*)d_in[3];
    const float* L1 = (const float*)d_in[4];
    const float* L2 = (const float*)d_in[5];
    const float* L3 = (const float*)d_in[6];
    float* out = (float*)d_out;

    const size_t BH = (size_t)BATCH * HID;
    float* ws   = (float*)d_ws;
    float* imgN = ws;                           // BATCH*IMGD
    float* labc = imgN + (size_t)BATCH * IMGD;  // 3*BH
    float* mems = labc + 3 * BH;                // 3*BH
    float* act  = mems + 3 * BH;                // BH

    // Deterministic zero state every call (traces live in d_out).
    hipMemsetAsync(d_out, 0, 3 * BH * sizeof(float), stream);
    hipMemsetAsync(mems, 0, 3 * BH * sizeof(float), stream);

    norm_kernel<<<BATCH, 256, 0, stream>>>(x, imgN);
    labc_kernel<<<(unsigned)(3 * BH / 256), 256, 0, stream>>>(x, L1, L2, L3, labc);

    const float* Ws[3] = {W1, W2, W3};
    const dim3 grid(HID / 256, BATCH / 64);
    for (int t = 0; t < 6; ++t) {
        for (int i = 0; i < 3; ++i) {
            if (i == 0) {
                gemm_lif_kernel<IMGD><<<grid, 256, 0, stream>>>(imgN, Ws[0],
                                                                labc, mems);
            } else {
                gemm_lif_kernel<HID><<<grid, 256, 0, stream>>>(act, Ws[i],
                                                               labc + (size_t)i * BH,
                                                               mems + (size_t)i * BH);
            }
            topk_lif_kernel<<<BATCH, 256, 0, stream>>>(mems + (size_t)i * BH,
                                                       act,
                                                       out + (size_t)i * BH);
        }
    }
}